// SDTWVL_26225070309733
// MI455X (gfx1250) — compile-verified
//
#include <hip/hip_runtime.h>
#include <hip/hip_bf16.h>

// SDTW-style variable-length loss, MI455X (gfx1250, wave32).
// Memory-bound gather+log+reduce: ~270MB traffic -> ~11.6us roofline @23.3TB/s
// (halved on average by the endframenum early-exit).
// Channel reduction via V_WMMA_F32_16X16X4_F32 (ones-matrix row-sum).
// Logs use raw v_log_f32 (log2) -- arguments are always normal/finite/positive
// (x*~1 + 1e-10, x in [0.01,0.99]) -- with ln2 folded into hoisted, row-masked
// coefficients so the hot loop is: 2 fma + 2 v_log_f32 + 1 sub + 2 mul per elem.

typedef __attribute__((ext_vector_type(2))) float v2f;
typedef __attribute__((ext_vector_type(8))) float v8f;

namespace {
constexpr float kPContent = 2.0f;
constexpr float kPEnd     = 0.2f;
constexpr float kEps      = 1e-10f;
constexpr float kS9       = 1.0f - 9.0f * kEps;   // == 1.0f in fp32, same as ref fp32 math
constexpr float kS2       = 1.0f - 2.0f * kEps;
constexpr float kLn2      = 0.69314718055994530942f;
}

__global__ __launch_bounds__(256) void sdtwvl_partial_kernel(
    const float* __restrict__ gen, const float* __restrict__ gt,
    const int* __restrict__ endframenum, const int* __restrict__ path,
    float* __restrict__ partials, int T, int Tg, int C)
{
  const int b    = blockIdx.x;
  const int wave = threadIdx.x >> 5;     // 8 waves per block
  const int lane = threadIdx.x & 31;
  const int half = lane >> 4;            // 0: K={0,1}, 1: K={2,3} of the A tile
  const int row  = lane & 15;            // M index within 16-row tile
  const int ntile = T >> 4;
  const int end   = endframenum[b];      // rows t in [0, end) contribute
  const float kContent = (2.0f * kPContent) / (float)(C - 1);
  const int kIters = C >> 2;             // 4 channels per WMMA K-step

  float waveTotal = 0.0f;

  const int tile = blockIdx.y * 8 + wave;   // wave-uniform
  const int t0   = tile << 4;
  if (tile < ntile && t0 < end) {           // wave-uniform skip past endframenum
    const int t      = t0 + row;
    const bool rowOn = (t < end);
    // Row mask and ln2 (log2 -> ln) folded into hoisted coefficients.
    const float kc2 = rowOn ? (kContent * kLn2) : 0.0f;  // content scale
    const float pe2 = rowOn ? (-kPEnd * kLn2)   : 0.0f;  // end-channel scale
    const int p      = path[(size_t)b * T + t];
    const float* genRow = gen + ((size_t)b * T  + t) * C;
    const float* gtRow  = gt  + ((size_t)b * Tg + p) * C;

    v8f acc = {0.f, 0.f, 0.f, 0.f, 0.f, 0.f, 0.f, 0.f};
    const v2f ones = {1.0f, 1.0f};          // B = all-ones => D rows = row sums of A

    // Hot loop: pure content path (never touches channel C-1).
    for (int k = 0; k < kIters - 1; ++k) {
      const int c0 = (k << 2) + (half << 1);
      // lanes L and L+16 jointly cover 16 contiguous bytes of the same row
      v2f g  = *(const v2f*)(genRow + c0);  // global_load_b64
      v2f gr = *(const v2f*)(gtRow  + c0);  // gathered row via path
      v2f a;
      a.x = kc2 * gr.x * (__builtin_amdgcn_logf(fmaf(gr.x, kS9, kEps)) -
                          __builtin_amdgcn_logf(fmaf(g.x,  kS9, kEps)));
      a.y = kc2 * gr.y * (__builtin_amdgcn_logf(fmaf(gr.y, kS9, kEps)) -
                          __builtin_amdgcn_logf(fmaf(g.y,  kS9, kEps)));
      // D(16x16) += A(16x4) * ones(4x16): accumulates per-row channel sums
      acc = __builtin_amdgcn_wmma_f32_16x16x4_f32(
          false, a, false, ones, (short)0, acc, false, false);
    }

    // Peeled last K-chunk: channels {C-4,C-3} (half 0) / {C-2,C-1} (half 1).
    {
      const int c0 = ((kIters - 1) << 2) + (half << 1);
      v2f g  = *(const v2f*)(genRow + c0);
      v2f gr = *(const v2f*)(gtRow  + c0);
      v2f a;
      a.x = kc2 * gr.x * (__builtin_amdgcn_logf(fmaf(gr.x, kS9, kEps)) -
                          __builtin_amdgcn_logf(fmaf(g.x,  kS9, kEps)));
      const bool isEnd = (half == 1);       // element c0+1 == C-1
      float a1 = isEnd ? fmaf(g.y,        kS2, kEps) : fmaf(gr.y, kS9, kEps);
      float a2 = isEnd ? fmaf(1.0f - g.y, kS2, kEps) : fmaf(g.y,  kS9, kEps);
      float l1 = __builtin_amdgcn_logf(a1);   // log2
      float l2 = __builtin_amdgcn_logf(a2);   // log2
      float vEnd = pe2 * fmaf(gr.y, l1, (1.0f - gr.y) * l2);  // end BCE (masked)
      float vCon = kc2 * gr.y * (l1 - l2);                    // content (masked)
      a.y = isEnd ? vEnd : vCon;
      acc = __builtin_amdgcn_wmma_f32_16x16x4_f32(
          false, a, false, ones, (short)0, acc, false, false);
    }

    // Column of D: lanes 0-15 VGPR r -> row r; lanes 16-31 VGPR r -> row r+8.
    float s = acc[0] + acc[1] + acc[2] + acc[3] + acc[4] + acc[5] + acc[6] + acc[7];
    s += __shfl_xor(s, 16, 32);            // rows 0-7 + rows 8-15 => tile total
    waveTotal = s;
  }

  __shared__ float wsum[8];
  if (lane == 0) wsum[wave] = waveTotal;
  __syncthreads();
  if (threadIdx.x == 0) {
    float tot = 0.0f;
#pragma unroll
    for (int i = 0; i < 8; ++i) tot += wsum[i];
    // Deterministic two-pass reduction: per-(b,chunk) partial into workspace
    partials[(size_t)b * gridDim.y + blockIdx.y] = tot;
  }
}

__global__ void sdtwvl_finalize_kernel(const float* __restrict__ partials,
                                       const int* __restrict__ endframenum,
                                       float* __restrict__ out,
                                       int B, int nchunks)
{
  const int b = blockIdx.x * blockDim.x + threadIdx.x;
  if (b < B) {
    float tot = 0.0f;
    for (int i = 0; i < nchunks; ++i) tot += partials[(size_t)b * nchunks + i];
    out[b]         = tot;                            // ans = cumsum at idx
    out[B + b]     = (float)(endframenum[b] - 1);    // idx
    out[2 * B + b] = 0.0f;                           // zeros
  }
}

extern "C" void kernel_launch(void* const* d_in, const int* in_sizes, int n_in,
                              void* d_out, int out_size, void* d_ws, size_t ws_size,
                              hipStream_t stream) {
  const float* gen        = (const float*)d_in[0];
  const float* gt         = (const float*)d_in[1];
  const int* endframenum  = (const int*)d_in[2];
  // d_in[3] = framelen: unused by the reference computation
  const int* path         = (const int*)d_in[4];

  const int B  = in_sizes[2];
  const int T  = in_sizes[4] / B;
  const int C  = in_sizes[0] / (B * T);
  const int Tg = in_sizes[1] / (B * C);

  const int ntile  = T / 16;              // T=2048 -> 128 tiles of 16 rows
  const int chunks = (ntile + 7) / 8;     // 8 waves/block -> 16 chunks

  float* partials = (float*)d_ws;         // needs B*chunks*4 bytes (4 KB)

  sdtwvl_partial_kernel<<<dim3(B, chunks), 256, 0, stream>>>(
      gen, gt, endframenum, path, partials, T, Tg, C);
  sdtwvl_finalize_kernel<<<dim3((B + 63) / 64), 64, 0, stream>>>(
      partials, endframenum, (float*)d_out, B, chunks);
}